// LBPLayer_35820027248912
// MI455X (gfx1250) — compile-verified
//
#include <hip/hip_runtime.h>
#include <cstdint>
#include <cstddef>

// LBP layer, faithful to the reference's transposed grid_sample convention:
//   out[b,h,w] = sum_i 2^i * [ bilinear(x[b]; row = clip(w+dy_i), col = clip(h+dx_i)) >= x[b,h,w] ]
// H = W = 1024, 24 points, radius 3, border-clamped.
//
// Strategy (MI455X): workload is gather/compare — not WMMA-shaped. It IS
// transposed-gather bound, so each 64x64 output tile stages its 72x72
// (halo=4) transposed sample neighborhood into LDS via CDNA5 async
// global->LDS DMA (ASYNCcnt), then all 24*4 gathers per pixel hit LDS
// (stride 73 => bank-conflict-free for lane-consecutive rows). Center
// reads / output writes are directly coalesced. 128 MB total HBM traffic
// (~5.5us floor at 23.3 TB/s); kernel is VALU/LDS-throughput bound.

#define HH    1024
#define WW    1024
#define TILE  64
#define HALO  4
#define AROWS (TILE + 2 * HALO)   // 72
#define ASTR  (AROWS + 1)         // 73: 73 % 64 == 9, conflict-free row stride
#define NPTS  24

typedef __attribute__((address_space(1))) int gint;   // global (prints as __device__ int*)
typedef __attribute__((address_space(3))) int lint;   // LDS    (prints as __shared__ int*)

#if __has_builtin(__builtin_amdgcn_global_load_async_to_lds_b32)
#define USE_ASYNC_LDS 1
#else
#define USE_ASYNC_LDS 0
#endif

__device__ __forceinline__ void wait_async_zero() {
#if __has_builtin(__builtin_amdgcn_s_wait_asynccnt)
  __builtin_amdgcn_s_wait_asynccnt(0);
#else
  asm volatile("s_wait_asynccnt 0" ::: "memory");
#endif
}

__global__ __launch_bounds__(256) void lbp_tile_kernel(const float* __restrict__ x,
                                                       float* __restrict__ out) {
  __shared__ float As[AROWS * ASTR];

  const int tid = threadIdx.x;
  const int w0  = blockIdx.x * TILE;
  const int h0  = blockIdx.y * TILE;
  const int b   = blockIdx.z;

  const float* __restrict__ src = x + (size_t)b * (HH * WW);

  // ---- Stage sample region into LDS: As[r][c] = x[b, clamp(w0-4+r), clamp(h0-4+c)]
  // Rows are contiguous in memory -> coalesced async DMA, no VGPR round trip.
  for (int idx = tid; idx < AROWS * AROWS; idx += 256) {
    const int r = idx / AROWS;
    const int c = idx - r * AROWS;
    int gy = w0 - HALO + r; gy = gy < 0 ? 0 : (gy > HH - 1 ? HH - 1 : gy);
    int gx = h0 - HALO + c; gx = gx < 0 ? 0 : (gx > WW - 1 ? WW - 1 : gx);
    const float* gp = src + (size_t)gy * WW + gx;
    float* lp = &As[r * ASTR + c];
#if USE_ASYNC_LDS
    __builtin_amdgcn_global_load_async_to_lds_b32((gint*)gp, (lint*)lp, 0, 0);
#else
    *lp = *gp;
#endif
  }
#if USE_ASYNC_LDS
  wait_async_zero();
#endif
  __syncthreads();

  // ---- Each thread: one w column, 16 h rows of the output tile.
  const int tx    = tid & (TILE - 1);
  const int ty    = tid >> 6;           // 0..3
  const int hbase = h0 + ty * 16;
  const int w     = w0 + tx;

  float cen[16], accv[16];
#pragma unroll
  for (int k = 0; k < 16; ++k) {
    cen[k]  = src[(size_t)(hbase + k) * WW + w];   // coalesced along tx
    accv[k] = 0.0f;
  }

  const int rowBase = -(w0 - HALO);   // clamped global row -> LDS row
  const int colBase = -(h0 - HALO);   // clamped global col -> LDS col

  float scale = 1.0f;
#pragma unroll 1                       // keep the 24-point loop rolled (I$)
  for (int i = 0; i < NPTS; ++i) {
    const float ang = (float)i * 0.26179939f;      // float32(2*pi/24)
    const float dxv = 3.0f * cosf(ang);
    const float dyv = 3.0f * sinf(ang);

    // w-side (sample ROW), shared by this thread's 16 outputs
    float py  = fminf(fmaxf((float)w + dyv, 0.0f), (float)(HH - 1));
    float y0f = floorf(py);
    float fy  = py - y0f;
    int   y0i = (int)y0f;
    int   y1i = (y0i + 1 > HH - 1) ? (HH - 1) : (y0i + 1);
    const float* __restrict__ ar0 = &As[(y0i + rowBase) * ASTR];
    const float* __restrict__ ar1 = &As[(y1i + rowBase) * ASTR];
    const float omfy = 1.0f - fy;

#pragma unroll
    for (int k = 0; k < 16; ++k) {
      // h-side (sample COLUMN)
      float px  = fminf(fmaxf((float)(hbase + k) + dxv, 0.0f), (float)(WW - 1));
      float x0f = floorf(px);
      float fx  = px - x0f;
      int   x0i = (int)x0f;
      int   x1i = (x0i + 1 > WW - 1) ? (WW - 1) : (x0i + 1);
      const int c0 = x0i + colBase;
      const int c1 = x1i + colBase;

      const float v00 = ar0[c0], v10 = ar0[c1];
      const float v01 = ar1[c0], v11 = ar1[c1];
      const float omfx = 1.0f - fx;
      const float s = omfx * omfy * v00 + fx * omfy * v10
                    + omfx * fy  * v01 + fx * fy  * v11;

      accv[k] += (s >= cen[k]) ? scale : 0.0f;
    }
    scale *= 2.0f;
  }

  float* __restrict__ dst = out + (size_t)b * (HH * WW);
#pragma unroll
  for (int k = 0; k < 16; ++k) {
    dst[(size_t)(hbase + k) * WW + w] = accv[k];   // coalesced along tx
  }
}

extern "C" void kernel_launch(void* const* d_in, const int* in_sizes, int n_in,
                              void* d_out, int out_size, void* d_ws, size_t ws_size,
                              hipStream_t stream) {
  (void)n_in; (void)out_size; (void)d_ws; (void)ws_size;
  const float* x  = (const float*)d_in[0];
  float* out      = (float*)d_out;
  const int batch = in_sizes[0] / (HH * WW);   // 16 for the reference shapes

  dim3 grid(WW / TILE, HH / TILE, batch);      // (16, 16, 16)
  lbp_tile_kernel<<<grid, 256, 0, stream>>>(x, out);
}